// TotalLossWithPriorAndRegularization_11802570129775
// MI455X (gfx1250) — compile-verified
//
#include <hip/hip_runtime.h>

typedef float v2f __attribute__((ext_vector_type(2)));
typedef float v8f __attribute__((ext_vector_type(8)));

#define B_ROWS 1048576
#define C_CLS  22
#define CLIPC  0.05f
#define EPSC   1e-8f
#define KL_EPSC 1e-6f

// Workspace layout (floats):
// [0..31]    sum_p per class (padded to 32)
// [32..63]   sum_p2 per class
// [64..1087] cross (32x32 Gram accumulation, padded from 22x22)
// [1088]     asym loss sum
#define WS_SUMP   0
#define WS_SUMP2  32
#define WS_CROSS  64
#define WS_ASYM   1088
#define WS_TOTAL  1089

__global__ void zero_ws_kernel(float* __restrict__ ws) {
    int i = blockIdx.x * blockDim.x + threadIdx.x;
    if (i < WS_TOTAL) ws[i] = 0.0f;
}

__device__ __forceinline__ float proc_elem(float x, float y, float& lossAcc) {
    const float LOG2E = 1.4426950408889634f;
    const float LN2   = 0.6931471805599453f;
    // sigmoid via native v_exp_f32 + v_rcp_f32 (avoid IEEE div sequence)
    float e  = __builtin_amdgcn_exp2f(-x * LOG2E);       // exp(-x)
    float s  = __builtin_amdgcn_rcpf(1.0f + e);          // 1/(1+exp(-x))
    float xs_neg = fminf(1.0f - s + CLIPC, 1.0f);
    float logpos = __builtin_amdgcn_logf(fmaxf(s, EPSC)) * LN2;
    float logneg = __builtin_amdgcn_logf(fmaxf(xs_neg, EPSC)) * LN2;
    float los = y * logpos + (1.0f - y) * logneg;
    float pt  = s * y + xs_neg * (1.0f - y);
    float omp = 1.0f - pt;
    float omp2 = omp * omp;
    float w = y * omp + (1.0f - y) * (omp2 * omp2);  // gamma_pos=1, gamma_neg=4
    lossAcc -= los * w;
    return s;
}

__global__ __launch_bounds__(256)
void fused_loss_kernel(const float* __restrict__ preds,
                       const float* __restrict__ targets,
                       float* __restrict__ ws) {
    __shared__ float s_cross[1024];
    __shared__ float s_sum_p[32];
    __shared__ float s_sum_p2[32];
    __shared__ float s_asym;

    for (int i = threadIdx.x; i < 1024; i += blockDim.x) s_cross[i] = 0.0f;
    if (threadIdx.x < 32) { s_sum_p[threadIdx.x] = 0.0f; s_sum_p2[threadIdx.x] = 0.0f; }
    if (threadIdx.x == 0) s_asym = 0.0f;
    __syncthreads();

    const int lane    = threadIdx.x & 31;
    const int wave    = threadIdx.x >> 5;
    const int wavesPerBlock = blockDim.x >> 5;
    const int gwave   = blockIdx.x * wavesPerBlock + wave;
    const int nwaves  = gridDim.x * wavesPerBlock;

    const int halfsel = lane >> 4;   // 0: rows b0,b0+1 ; 1: rows b0+2,b0+3
    const int lcol    = lane & 15;

    v8f acc00 = {}; v8f acc01 = {}; v8f acc10 = {}; v8f acc11 = {};
    float sp0 = 0.0f, sp1 = 0.0f, sq0 = 0.0f, sq1 = 0.0f;
    float lossAcc = 0.0f;

#pragma unroll 2
    for (int b0 = gwave * 4; b0 < B_ROWS; b0 += nwaves * 4) {
        const int r0 = b0 + halfsel * 2;
        const int r1 = r0 + 1;

        // Tile column group 0: column lcol (always < 22)
        float s00, s01;
        {
            const int c = lcol;
            float x0 = preds[r0 * C_CLS + c];
            float x1 = preds[r1 * C_CLS + c];
            float y0 = targets[r0 * C_CLS + c];
            float y1 = targets[r1 * C_CLS + c];
            s00 = proc_elem(x0, y0, lossAcc);
            s01 = proc_elem(x1, y1, lossAcc);
            sp0 += s00 + s01;
            sq0 += s00 * s00 + s01 * s01;
        }
        // Tile column group 1: column 16+lcol (valid while < 22; pad prob = 0)
        float s10 = 0.0f, s11 = 0.0f;
        {
            const int c = 16 + lcol;
            if (c < C_CLS) {
                float x0 = preds[r0 * C_CLS + c];
                float x1 = preds[r1 * C_CLS + c];
                float y0 = targets[r0 * C_CLS + c];
                float y1 = targets[r1 * C_CLS + c];
                s10 = proc_elem(x0, y0, lossAcc);
                s11 = proc_elem(x1, y1, lossAcc);
                sp1 += s10 + s11;
                sq1 += s10 * s10 + s11 * s11;
            }
        }

        // Same lane data serves as both A (16x4 of P^T) and B (4x16 of P)
        v2f t0 = { s00, s01 };
        v2f t1 = { s10, s11 };

        // Gram tile update: acc[ti][tj] += A(ti) x B(tj)   (D = A*B + C)
        acc00 = __builtin_amdgcn_wmma_f32_16x16x4_f32(false, t0, false, t0, (short)0, acc00, false, false);
        acc01 = __builtin_amdgcn_wmma_f32_16x16x4_f32(false, t0, false, t1, (short)0, acc01, false, false);
        acc10 = __builtin_amdgcn_wmma_f32_16x16x4_f32(false, t1, false, t0, (short)0, acc10, false, false);
        acc11 = __builtin_amdgcn_wmma_f32_16x16x4_f32(false, t1, false, t1, (short)0, acc11, false, false);
    }

    // ---- block-local reduction into LDS ----
    atomicAdd(&s_asym, lossAcc);
    atomicAdd(&s_sum_p[lcol],        sp0);
    atomicAdd(&s_sum_p[16 + lcol],   sp1);
    atomicAdd(&s_sum_p2[lcol],       sq0);
    atomicAdd(&s_sum_p2[16 + lcol],  sq1);

    // C/D layout: VGPR v, lanes 0-15 -> M=v, N=lane ; lanes 16-31 -> M=v+8, N=lane-16
#pragma unroll
    for (int v = 0; v < 8; ++v) {
        const int mi = v + halfsel * 8;
        atomicAdd(&s_cross[(mi)      * 32 + lcol],        acc00[v]);
        atomicAdd(&s_cross[(mi)      * 32 + 16 + lcol],   acc01[v]);
        atomicAdd(&s_cross[(16 + mi) * 32 + lcol],        acc10[v]);
        atomicAdd(&s_cross[(16 + mi) * 32 + 16 + lcol],   acc11[v]);
    }
    __syncthreads();

    // ---- one global atomic sweep per block ----
    for (int i = threadIdx.x; i < 1024; i += blockDim.x)
        atomicAdd(&ws[WS_CROSS + i], s_cross[i]);
    if (threadIdx.x < 32) {
        atomicAdd(&ws[WS_SUMP  + threadIdx.x], s_sum_p[threadIdx.x]);
        atomicAdd(&ws[WS_SUMP2 + threadIdx.x], s_sum_p2[threadIdx.x]);
    }
    if (threadIdx.x == 0) atomicAdd(&ws[WS_ASYM], s_asym);
}

__global__ void finalize_kernel(const float* __restrict__ ws,
                                const float* __restrict__ sim,
                                const float* __restrict__ prior,
                                float* __restrict__ out) {
    if (threadIdx.x != 0 || blockIdx.x != 0) return;
    const float LN2 = 0.6931471805599453f;
    const float invB = 1.0f / (float)B_ROWS;

    float reg = 0.0f;
    for (int i = 0; i < C_CLS; ++i) {
        const float m2i = ws[WS_SUMP2 + i] * invB;
        for (int j = 0; j < C_CLS; ++j) {
            const float m2j = ws[WS_SUMP2 + j] * invB;
            const float cr  = ws[WS_CROSS + i * 32 + j] * invB;
            reg += sim[i * C_CLS + j] * (m2i + m2j - 2.0f * cr);
        }
    }

    float kl = 0.0f;
    for (int c = 0; c < C_CLS; ++c) {
        const float pm = ws[WS_SUMP + c] * invB;
        const float p  = prior[c];
        kl += p * (__builtin_amdgcn_logf((p + KL_EPSC) / (pm + KL_EPSC)) * LN2);
    }

    out[0] = ws[WS_ASYM] + 1.0f * reg + 0.5f * kl;
}

extern "C" void kernel_launch(void* const* d_in, const int* in_sizes, int n_in,
                              void* d_out, int out_size, void* d_ws, size_t ws_size,
                              hipStream_t stream) {
    const float* preds   = (const float*)d_in[0];
    const float* targets = (const float*)d_in[1];
    const float* sim     = (const float*)d_in[2];
    const float* prior   = (const float*)d_in[3];
    float* ws  = (float*)d_ws;
    float* out = (float*)d_out;

    // deterministic zero-init of workspace on-stream every launch
    zero_ws_kernel<<<(WS_TOTAL + 255) / 256, 256, 0, stream>>>(ws);

    // 512 blocks x 8 waves = 4096 waves; each wave eats 4 rows/iter -> 64 iters
    fused_loss_kernel<<<512, 256, 0, stream>>>(preds, targets, ws);

    finalize_kernel<<<1, 32, 0, stream>>>(ws, sim, prior, out);
}